// InvoMLP_52664888983874
// MI455X (gfx1250) — compile-verified
//
#include <hip/hip_runtime.h>
#include <hip/hip_bf16.h>
#include <math.h>

// ---------------- problem constants ----------------
#define BB 16
#define HH 56
#define WW 56
#define CC 256
#define C4 64
#define GG 8
#define OGEN 72
#define NPOS 50176      // BB*HH*WW
#define HWSZ 3136       // HH*WW

typedef float v2f __attribute__((ext_vector_type(2)));
typedef float v8f __attribute__((ext_vector_type(8)));

#define WMMA_F32(acc, a, b) \
  acc = __builtin_amdgcn_wmma_f32_16x16x4_f32(false, (a), false, (b), (short)0, (acc), false, false)

__device__ __forceinline__ v2f ld2(const float* p) { return *(const v2f*)p; }

// ---------------- workspace layout (float offsets) ----------------
#define OFF_T      ((size_t)0)                       // 50176*64
#define OFF_WGEN   (OFF_T + (size_t)NPOS * C4)       // 50176*72
#define OFF_HW     (OFF_WGEN + (size_t)NPOS * OGEN)  // 50176*256
#define OFF_CB     (OFF_HW + (size_t)NPOS * CC)      // 50176*256
#define OFF_STATS  (OFF_CB + (size_t)NPOS * CC)      // 128 (sum, sumsq)
#define OFF_AACC   (OFF_STATS + 128)                 // 16*256
#define OFF_SCALE  (OFF_AACC + 4096)                 // 64
#define OFF_SHIFT  (OFF_SCALE + 64)                  // 64
#define OFF_A0     (OFF_SHIFT + 64)                  // 16*256
#define OFF_A1     (OFF_A0 + 4096)                   // 16*256

// ---------------- utility ----------------
__global__ void zero_kernel(float* __restrict__ p, int n) {
  int i = blockIdx.x * blockDim.x + threadIdx.x;
  if (i < n) p[i] = 0.0f;
}

// ---------------- conv1: T[N,64] = X[N,256] @ W1[64,256]^T + b ----------------
// 8 waves; wave tile = 32 rows x 64 cols (2 M-tiles x 4 N-tiles = 8 accums)
__global__ __launch_bounds__(256) void conv1_gemm(
    const float* __restrict__ X, const float* __restrict__ W1,
    const float* __restrict__ B1, float* __restrict__ T) {
  const int tid  = threadIdx.x;
  const int wave = tid >> 5, lane = tid & 31;
  const int hp   = lane >> 4, l = lane & 15;
  const int m0   = blockIdx.x * 256 + wave * 32;

  v8f acc[2][4] = {};
  const float* arow0 = X + (size_t)(m0 + l) * CC;
  const float* arow1 = arow0 + (size_t)16 * CC;
  const float* b0r = W1 + (size_t)( 0 + l) * CC;
  const float* b1r = W1 + (size_t)(16 + l) * CC;
  const float* b2r = W1 + (size_t)(32 + l) * CC;
  const float* b3r = W1 + (size_t)(48 + l) * CC;
  for (int kb = 0; kb < 64; ++kb) {
    const int k = kb * 4 + hp * 2;
    v2f a0 = ld2(arow0 + k);
    v2f a1 = ld2(arow1 + k);
    v2f b0 = ld2(b0r + k), b1 = ld2(b1r + k);
    v2f b2 = ld2(b2r + k), b3 = ld2(b3r + k);
    WMMA_F32(acc[0][0], a0, b0);
    WMMA_F32(acc[0][1], a0, b1);
    WMMA_F32(acc[0][2], a0, b2);
    WMMA_F32(acc[0][3], a0, b3);
    WMMA_F32(acc[1][0], a1, b0);
    WMMA_F32(acc[1][1], a1, b1);
    WMMA_F32(acc[1][2], a1, b2);
    WMMA_F32(acc[1][3], a1, b3);
  }
  const float bi0 = B1[ 0 + l], bi1 = B1[16 + l];
  const float bi2 = B1[32 + l], bi3 = B1[48 + l];
  #pragma unroll
  for (int mi = 0; mi < 2; ++mi) {
    #pragma unroll
    for (int i = 0; i < 8; ++i) {
      float* orow = T + (size_t)(m0 + mi * 16 + i + hp * 8) * C4;
      orow[ 0 + l] = acc[mi][0][i] + bi0;
      orow[16 + l] = acc[mi][1][i] + bi1;
      orow[32 + l] = acc[mi][2][i] + bi2;
      orow[48 + l] = acc[mi][3][i] + bi3;
    }
  }
}

// ---------------- BN stats: per-channel sum / sumsq over 50176 rows ----------------
__global__ __launch_bounds__(256) void bn_stats_kernel(
    const float* __restrict__ T, float* __restrict__ stats) {
  const int c  = threadIdx.x & 63;
  const int rg = threadIdx.x >> 6;   // 0..3
  const int r0 = blockIdx.x * 392;   // 50176 / 128 blocks
  float s = 0.0f, sq = 0.0f;
  for (int r = rg; r < 392; r += 4) {
    float v = T[(size_t)(r0 + r) * C4 + c];
    s += v; sq += v * v;
  }
  __shared__ float ls[256], lq[256];
  ls[threadIdx.x] = s; lq[threadIdx.x] = sq;
  __syncthreads();
  if (rg == 0) {
    s  = ls[c] + ls[c + 64] + ls[c + 128] + ls[c + 192];
    sq = lq[c] + lq[c + 64] + lq[c + 128] + lq[c + 192];
    atomicAdd(&stats[c], s);
    atomicAdd(&stats[64 + c], sq);
  }
}

__global__ void bn_finalize(const float* __restrict__ stats,
                            const float* __restrict__ gamma,
                            const float* __restrict__ beta,
                            float* __restrict__ scale, float* __restrict__ shift) {
  const int c = threadIdx.x;  // 64 threads
  const float inv_n = 1.0f / (float)NPOS;
  const float mean = stats[c] * inv_n;
  const float var  = stats[64 + c] * inv_n - mean * mean;
  const float is   = rsqrtf(var + 1e-5f);
  const float sc   = gamma[c] * is;
  scale[c] = sc;
  shift[c] = beta[c] - mean * sc;
}

// ---------------- conv2: WG[N,72] = relu(bn(T)) @ W2[72,64]^T + b ----------------
// wave tile = 32 rows x 80 cols (2 M-tiles x 5 N-tiles, masked at 72)
__global__ __launch_bounds__(256) void conv2_gemm(
    const float* __restrict__ T, const float* __restrict__ scale,
    const float* __restrict__ shift, const float* __restrict__ W2,
    const float* __restrict__ B2, float* __restrict__ WG) {
  const int tid  = threadIdx.x;
  const int wave = tid >> 5, lane = tid & 31;
  const int hp   = lane >> 4, l = lane & 15;
  const int m0   = blockIdx.x * 256 + wave * 32;

  v8f acc[2][5] = {};
  const float* arow0 = T + (size_t)(m0 + l) * C4;
  const float* arow1 = arow0 + (size_t)16 * C4;
  for (int kb = 0; kb < 16; ++kb) {
    const int k = kb * 4 + hp * 2;
    v2f sc = ld2(scale + k);
    v2f sh = ld2(shift + k);
    v2f t0 = ld2(arow0 + k);
    v2f t1 = ld2(arow1 + k);
    v2f a0, a1;
    a0.x = fmaxf(sc.x * t0.x + sh.x, 0.0f);
    a0.y = fmaxf(sc.y * t0.y + sh.y, 0.0f);
    a1.x = fmaxf(sc.x * t1.x + sh.x, 0.0f);
    a1.y = fmaxf(sc.y * t1.y + sh.y, 0.0f);
    #pragma unroll
    for (int j = 0; j < 5; ++j) {
      const int o = j * 16 + l;
      v2f b; b.x = 0.0f; b.y = 0.0f;
      if (o < OGEN) b = ld2(W2 + (size_t)o * C4 + k);
      WMMA_F32(acc[0][j], a0, b);
      WMMA_F32(acc[1][j], a1, b);
    }
  }
  #pragma unroll
  for (int j = 0; j < 5; ++j) {
    const int n = j * 16 + l;
    if (n < OGEN) {
      const float bias = B2[n];
      #pragma unroll
      for (int mi = 0; mi < 2; ++mi)
        #pragma unroll
        for (int i = 0; i < 8; ++i)
          WG[(size_t)(m0 + mi * 16 + i + hp * 8) * OGEN + n] = acc[mi][j][i] + bias;
    }
  }
}

// ---------------- involution: HWo[pos, c] = sum_kk WG[pos, g*9+kk] * X[nbr(pos,kk), c] ----------------
__global__ __launch_bounds__(256) void involution_kernel(
    const float* __restrict__ X, const float* __restrict__ WG,
    float* __restrict__ HWo) {
  const int pos = blockIdx.x;       // 0..NPOS-1
  const int c   = threadIdx.x;      // 0..255
  const int b   = pos / HWSZ;
  const int hwi = pos - b * HWSZ;
  const int h   = hwi / WW;
  const int w   = hwi - h * WW;
  const int g   = c >> 5;           // C/G = 32 channels per group
  const float* wk = WG + (size_t)pos * OGEN + g * 9;
  float acc = 0.0f;
  #pragma unroll
  for (int i = 0; i < 3; ++i) {
    const int hh = h + i - 1;
    if (hh < 0 || hh >= HH) continue;
    #pragma unroll
    for (int j = 0; j < 3; ++j) {
      const int wc = w + j - 1;
      if (wc < 0 || wc >= WW) continue;
      acc += wk[i * 3 + j] * X[(((size_t)b * HH + hh) * WW + wc) * CC + c];
    }
  }
  HWo[(size_t)pos * CC + c] = acc;
}

// ---------------- mlp_c: Cb[N,256] = X @ Wm[256,256]^T ----------------
// 8 waves = 2 m-groups x 4 n-groups; wave tile = 32x64 (8 accums)
__global__ __launch_bounds__(256) void mlpc_gemm(
    const float* __restrict__ X, const float* __restrict__ Wm,
    float* __restrict__ Cb) {
  const int tid  = threadIdx.x;
  const int wave = tid >> 5, lane = tid & 31;
  const int hp   = lane >> 4, l = lane & 15;
  const int m0   = blockIdx.x * 64 + (wave >> 2) * 32;
  const int n0   = (wave & 3) * 64;

  v8f acc[2][4] = {};
  const float* arow0 = X + (size_t)(m0 + l) * CC;
  const float* arow1 = arow0 + (size_t)16 * CC;
  const float* b0r = Wm + (size_t)(n0 +  0 + l) * CC;
  const float* b1r = Wm + (size_t)(n0 + 16 + l) * CC;
  const float* b2r = Wm + (size_t)(n0 + 32 + l) * CC;
  const float* b3r = Wm + (size_t)(n0 + 48 + l) * CC;
  for (int kb = 0; kb < 64; ++kb) {
    const int k = kb * 4 + hp * 2;
    v2f a0 = ld2(arow0 + k);
    v2f a1 = ld2(arow1 + k);
    v2f b0 = ld2(b0r + k), b1 = ld2(b1r + k);
    v2f b2 = ld2(b2r + k), b3 = ld2(b3r + k);
    WMMA_F32(acc[0][0], a0, b0);
    WMMA_F32(acc[0][1], a0, b1);
    WMMA_F32(acc[0][2], a0, b2);
    WMMA_F32(acc[0][3], a0, b3);
    WMMA_F32(acc[1][0], a1, b0);
    WMMA_F32(acc[1][1], a1, b1);
    WMMA_F32(acc[1][2], a1, b2);
    WMMA_F32(acc[1][3], a1, b3);
  }
  #pragma unroll
  for (int mi = 0; mi < 2; ++mi) {
    #pragma unroll
    for (int i = 0; i < 8; ++i) {
      float* orow = Cb + (size_t)(m0 + mi * 16 + i + hp * 8) * CC + n0;
      orow[ 0 + l] = acc[mi][0][i];
      orow[16 + l] = acc[mi][1][i];
      orow[32 + l] = acc[mi][2][i];
      orow[48 + l] = acc[mi][3][i];
    }
  }
}

// ---------------- spatial mean: AACC[b,c] += sum_pos (HW + Cb) ----------------
__global__ __launch_bounds__(256) void mean_reduce(
    const float* __restrict__ HWb, const float* __restrict__ Cb,
    float* __restrict__ AACC) {
  const int b     = blockIdx.x >> 3;
  const int chunk = blockIdx.x & 7;
  const int c     = threadIdx.x;
  const size_t p0 = (size_t)b * HWSZ + chunk * 392;
  float s = 0.0f;
  for (int r = 0; r < 392; ++r) {
    const size_t idx = (p0 + r) * CC + c;
    s += HWb[idx] + Cb[idx];
  }
  atomicAdd(&AACC[b * CC + c], s);
}

// ---------------- tiny MLP: mean -> fc1+gelu -> fc2 -> pairwise softmax ----------------
__global__ __launch_bounds__(256) void mlp_small(
    const float* __restrict__ AACC,
    const float* __restrict__ F1W, const float* __restrict__ F1B,
    const float* __restrict__ F2W, const float* __restrict__ F2B,
    float* __restrict__ A0, float* __restrict__ A1) {
  __shared__ float a_s[BB * CC];   // 16 KB
  __shared__ float h_s[BB * C4];   // 4 KB
  const int tid = threadIdx.x;
  for (int i = tid; i < BB * CC; i += 256) a_s[i] = AACC[i] * (1.0f / (float)HWSZ);
  __syncthreads();
  for (int o = tid; o < BB * C4; o += 256) {
    const int b = o >> 6, j = o & 63;
    float s = F1B[j];
    const float* ar = a_s + b * CC;
    const float* wr = F1W + (size_t)j * CC;
    for (int k = 0; k < CC; ++k) s += ar[k] * wr[k];
    h_s[o] = 0.5f * s * (1.0f + erff(s * 0.70710678118654752f));   // exact gelu
  }
  __syncthreads();
  for (int p = tid; p < BB * CC; p += 256) {
    const int b = p >> 8, cch = p & 255;
    const float* hr = h_s + b * C4;
    const float* w0 = F2W + (size_t)(2 * cch) * C4;
    const float* w1 = w0 + C4;
    float s0 = F2B[2 * cch], s1 = F2B[2 * cch + 1];
    for (int k = 0; k < C4; ++k) { s0 += hr[k] * w0[k]; s1 += hr[k] * w1[k]; }
    const float m  = fmaxf(s0, s1);
    const float e0 = expf(s0 - m), e1 = expf(s1 - m);
    const float inv = 1.0f / (e0 + e1);
    A0[p] = e0 * inv;
    A1[p] = e1 * inv;
  }
}

// ---------------- final: OUT = (HW*a0 + Cb*a1) @ Wp[256,256]^T + bp ----------------
// wave tile = 32x64; both M-tiles share one batch image (32 | 3136) -> shared gate loads
__global__ __launch_bounds__(256) void final_gemm(
    const float* __restrict__ HWb, const float* __restrict__ Cb,
    const float* __restrict__ A0, const float* __restrict__ A1,
    const float* __restrict__ Wp, const float* __restrict__ Bp,
    float* __restrict__ OUT) {
  const int tid  = threadIdx.x;
  const int wave = tid >> 5, lane = tid & 31;
  const int hp   = lane >> 4, l = lane & 15;
  const int m0   = blockIdx.x * 64 + (wave >> 2) * 32;
  const int n0   = (wave & 3) * 64;

  const int row  = m0 + l;
  const int bidx = row / HWSZ;   // same for row and row+16 (32 divides 3136)
  const float* hrow0 = HWb + (size_t)row * CC;
  const float* hrow1 = hrow0 + (size_t)16 * CC;
  const float* crow0 = Cb  + (size_t)row * CC;
  const float* crow1 = crow0 + (size_t)16 * CC;
  const float* a0r   = A0 + (size_t)bidx * CC;
  const float* a1r   = A1 + (size_t)bidx * CC;

  v8f acc[2][4] = {};
  const float* b0r = Wp + (size_t)(n0 +  0 + l) * CC;
  const float* b1r = Wp + (size_t)(n0 + 16 + l) * CC;
  const float* b2r = Wp + (size_t)(n0 + 32 + l) * CC;
  const float* b3r = Wp + (size_t)(n0 + 48 + l) * CC;
  for (int kb = 0; kb < 64; ++kb) {
    const int k = kb * 4 + hp * 2;
    v2f s0 = ld2(a0r + k);
    v2f s1 = ld2(a1r + k);
    v2f a0 = ld2(hrow0 + k) * s0 + ld2(crow0 + k) * s1;
    v2f a1 = ld2(hrow1 + k) * s0 + ld2(crow1 + k) * s1;
    v2f b0 = ld2(b0r + k), b1 = ld2(b1r + k);
    v2f b2 = ld2(b2r + k), b3 = ld2(b3r + k);
    WMMA_F32(acc[0][0], a0, b0);
    WMMA_F32(acc[0][1], a0, b1);
    WMMA_F32(acc[0][2], a0, b2);
    WMMA_F32(acc[0][3], a0, b3);
    WMMA_F32(acc[1][0], a1, b0);
    WMMA_F32(acc[1][1], a1, b1);
    WMMA_F32(acc[1][2], a1, b2);
    WMMA_F32(acc[1][3], a1, b3);
  }
  const float bi0 = Bp[n0 +  0 + l], bi1 = Bp[n0 + 16 + l];
  const float bi2 = Bp[n0 + 32 + l], bi3 = Bp[n0 + 48 + l];
  #pragma unroll
  for (int mi = 0; mi < 2; ++mi) {
    #pragma unroll
    for (int i = 0; i < 8; ++i) {
      float* orow = OUT + (size_t)(m0 + mi * 16 + i + hp * 8) * CC + n0;
      orow[ 0 + l] = acc[mi][0][i] + bi0;
      orow[16 + l] = acc[mi][1][i] + bi1;
      orow[32 + l] = acc[mi][2][i] + bi2;
      orow[48 + l] = acc[mi][3][i] + bi3;
    }
  }
}

// ---------------- launcher ----------------
extern "C" void kernel_launch(void* const* d_in, const int* in_sizes, int n_in,
                              void* d_out, int out_size, void* d_ws, size_t ws_size,
                              hipStream_t stream) {
  const float* x       = (const float*)d_in[0];
  const float* conv1_w = (const float*)d_in[1];
  const float* conv1_b = (const float*)d_in[2];
  const float* bn_g    = (const float*)d_in[3];
  const float* bn_b    = (const float*)d_in[4];
  const float* conv2_w = (const float*)d_in[5];
  const float* conv2_b = (const float*)d_in[6];
  const float* mlp_c_w = (const float*)d_in[7];
  const float* fc1_w   = (const float*)d_in[8];
  const float* fc1_b   = (const float*)d_in[9];
  const float* fc2_w   = (const float*)d_in[10];
  const float* fc2_b   = (const float*)d_in[11];
  const float* proj_w  = (const float*)d_in[12];
  const float* proj_b  = (const float*)d_in[13];
  float* out = (float*)d_out;
  float* ws  = (float*)d_ws;

  float* T     = ws + OFF_T;
  float* WG    = ws + OFF_WGEN;
  float* HWb   = ws + OFF_HW;
  float* Cb    = ws + OFF_CB;
  float* STATS = ws + OFF_STATS;
  float* AACC  = ws + OFF_AACC;
  float* SCALE = ws + OFF_SCALE;
  float* SHIFT = ws + OFF_SHIFT;
  float* A0    = ws + OFF_A0;
  float* A1    = ws + OFF_A1;

  // stats(128) + a_acc(4096) are contiguous; zero for atomics
  zero_kernel<<<17, 256, 0, stream>>>(STATS, 128 + 4096);

  conv1_gemm<<<NPOS / 256, 256, 0, stream>>>(x, conv1_w, conv1_b, T);
  bn_stats_kernel<<<128, 256, 0, stream>>>(T, STATS);
  bn_finalize<<<1, 64, 0, stream>>>(STATS, bn_g, bn_b, SCALE, SHIFT);
  conv2_gemm<<<NPOS / 256, 256, 0, stream>>>(T, SCALE, SHIFT, conv2_w, conv2_b, WG);
  involution_kernel<<<NPOS, 256, 0, stream>>>(x, WG, HWb);
  mlpc_gemm<<<NPOS / 64, 256, 0, stream>>>(x, mlp_c_w, Cb);
  mean_reduce<<<BB * 8, 256, 0, stream>>>(HWb, Cb, AACC);
  mlp_small<<<1, 256, 0, stream>>>(AACC, fc1_w, fc1_b, fc2_w, fc2_b, A0, A1);
  final_gemm<<<NPOS / 64, 256, 0, stream>>>(HWb, Cb, A0, A1, proj_w, proj_b, out);
}